// GraphTransformerLayer_82497731822092
// MI455X (gfx1250) — compile-verified
//
#include <hip/hip_runtime.h>

#define D      128
#define HH     8
#define DPH    16
#define EPSV   1e-5f
#define MT     5          // M-tiles (of 16 rows) per block: 80 rows, N = 625*80
#define LDA    (D + 4)    // padded LDS row stride (bank-conflict free: 132%64=4)

typedef __attribute__((ext_vector_type(2))) float v2f;
typedef __attribute__((ext_vector_type(8))) float v8f;
typedef int v4i __attribute__((vector_size(4 * sizeof(int))));

#if __has_builtin(__builtin_amdgcn_global_load_async_to_lds_b128) && \
    __has_builtin(__builtin_amdgcn_s_wait_asynccnt)
#define HAVE_ASYNC_LDS 1
#else
#define HAVE_ASYNC_LDS 0
#endif

// 16B global -> LDS copy, async (CDNA5 GLOBAL_LOAD_ASYNC_TO_LDS_B128) if available.
__device__ __forceinline__ void copy16(float* dst_lds, const float* src_glob) {
#if HAVE_ASYNC_LDS
  __builtin_amdgcn_global_load_async_to_lds_b128(
      (__attribute__((address_space(1))) v4i*)src_glob,
      (__attribute__((address_space(3))) v4i*)dst_lds, 0, 0);
#else
  *(float4*)dst_lds = *(const float4*)src_glob;
#endif
}

__device__ __forceinline__ void copy_fence() {
#if HAVE_ASYNC_LDS
  __builtin_amdgcn_s_wait_asynccnt(0);
#endif
  __syncthreads();
}

// ---------------------------------------------------------------------------
// WMMA f32 16x16x4, MT vertical 16x16 tiles sharing each B fragment.
// A-tile: (MT*16) x kdim, row stride LDA, in LDS. B: kdim x ncols row-major.
// Fragment layouts per CDNA5 ISA 7.12.2:
//   A (16x4 f32): lanes 0-15 hold M=lane, K={0,1}; lanes 16-31 hold K={2,3}
//   B (4x16 f32): lanes 0-15 hold N=lane, K={0,1}; lanes 16-31 hold K={2,3}
//   C/D (16x16 f32): VGPR r, lanes 0-15 -> M=r; lanes 16-31 -> M=8+r; N=lane&15
// ---------------------------------------------------------------------------
__device__ __forceinline__ void wmma_mt(const float* __restrict__ Atile, int kdim,
                                        const float* __restrict__ B, int ldb,
                                        int col0, v8f* c) {
  const int lane = threadIdx.x & 31;
  const int half = lane >> 4;
  const int lm   = lane & 15;
  const float* ar = Atile + lm * LDA + (half << 1);
  const float* br = B + (half << 1) * ldb + col0 + lm;
  for (int kk = 0; kk < kdim; kk += 4) {
    v2f b;
    b.x = br[kk * ldb];
    b.y = br[(kk + 1) * ldb];
#pragma unroll
    for (int m = 0; m < MT; ++m) {
      v2f a;
      a.x = ar[m * 16 * LDA + kk];
      a.y = ar[m * 16 * LDA + kk + 1];
      c[m] = __builtin_amdgcn_wmma_f32_16x16x4_f32(false, a, false, b, (short)0,
                                                   c[m], false, false);
    }
  }
}

// Cooperative copy of an 80 x D tile (row-major, row stride src_ld) into padded LDS.
__device__ __forceinline__ void load_atile(float* lds, const float* src,
                                           int src_ld) {
  for (int i = threadIdx.x; i < MT * 16 * (D / 4); i += blockDim.x) {
    const int r  = i >> 5;          // D/4 == 32 float4 per row
    const int c4 = (i & 31) << 2;
    copy16(lds + r * LDA + c4, src + (size_t)r * src_ld + c4);
  }
  copy_fence();
}

// ---------------------------------------------------------------------------
// Stage 1: Q = x@Wq, K = x@Wk, V = x@Wv
// ---------------------------------------------------------------------------
__global__ void k_qkv(const float* __restrict__ x,
                      const float* __restrict__ Wq, const float* __restrict__ Wk,
                      const float* __restrict__ Wv,
                      float* __restrict__ Q, float* __restrict__ K,
                      float* __restrict__ V) {
  __shared__ float a[MT * 16 * LDA];
  const int row0 = blockIdx.x * (MT * 16);
  load_atile(a, x + (size_t)row0 * D, D);

  const int wave = threadIdx.x >> 5;
  const int col0 = wave * 16;
  const int lane = threadIdx.x & 31, half = lane >> 4, lm = lane & 15;
  const float* Ws[3] = {Wq, Wk, Wv};
  float* Os[3] = {Q, K, V};
  for (int w = 0; w < 3; ++w) {
    v8f c[MT] = {};
    wmma_mt(a, D, Ws[w], D, col0, c);
#pragma unroll
    for (int m = 0; m < MT; ++m) {
      float* o = Os[w] + (size_t)(row0 + m * 16 + (half << 3)) * D + col0 + lm;
      for (int r = 0; r < 8; ++r) o[(size_t)r * D] = c[m][r];
    }
  }
}

// ---------------------------------------------------------------------------
// Stage 2: per-(edge,head) score -> exp -> scatter-add denominator.
// Scores are clipped to [-5,5], so exp() is safe without segment-max
// (alpha is invariant to the max shift).
// ---------------------------------------------------------------------------
__global__ void k_scores(const float* __restrict__ Q, const float* __restrict__ K,
                         const long long* __restrict__ src,
                         const long long* __restrict__ dst,
                         float* __restrict__ ex, float* __restrict__ denom, int E) {
  int idx = blockIdx.x * blockDim.x + threadIdx.x;
  if (idx >= E * HH) return;
  const int e = idx >> 3, h = idx & 7;
  const long long s = src[e], d = dst[e];
  const float4* q = (const float4*)(Q + (size_t)d * D + h * DPH);
  const float4* k = (const float4*)(K + (size_t)s * D + h * DPH);
  float acc = 0.f;
  for (int i = 0; i < 4; ++i) {
    float4 qa = q[i], ka = k[i];
    acc += qa.x * ka.x + qa.y * ka.y + qa.z * ka.z + qa.w * ka.w;
  }
  float sc = acc * 0.25f;                       // / sqrt(16)
  sc = fminf(5.f, fmaxf(-5.f, sc));
  float ev = __expf(sc);
  ex[idx] = ev;
  atomicAdd(denom + (size_t)d * HH + h, ev);
}

// ---------------------------------------------------------------------------
// Stage 3: message aggregation; atomics land in the 192MB L2 (att is 25.6MB).
// ---------------------------------------------------------------------------
__global__ void k_aggregate(const float* __restrict__ V, const float* __restrict__ ex,
                            const float* __restrict__ denom,
                            const long long* __restrict__ src,
                            const long long* __restrict__ dst,
                            float* __restrict__ att, int E) {
  int tid = blockIdx.x * blockDim.x + threadIdx.x;
  if (tid >= E * D) return;
  const int e = tid >> 7;
  const int d = tid & (D - 1);
  const int h = d >> 4;
  const long long s = src[e], dd = dst[e];
  const float alpha = ex[(size_t)e * HH + h] / denom[(size_t)dd * HH + h];
  atomicAdd(att + (size_t)dd * D + d, V[(size_t)s * D + d] * alpha);
}

// ---------------------------------------------------------------------------
// Stage 4: h1 = x + att @ Wo + bo
// ---------------------------------------------------------------------------
__global__ void k_oproj(const float* __restrict__ att, const float* __restrict__ Wo,
                        const float* __restrict__ bo, const float* __restrict__ x,
                        float* __restrict__ h1) {
  __shared__ float a[MT * 16 * LDA];
  const int row0 = blockIdx.x * (MT * 16);
  load_atile(a, att + (size_t)row0 * D, D);

  const int wave = threadIdx.x >> 5;
  const int col0 = wave * 16;
  const int lane = threadIdx.x & 31, half = lane >> 4, lm = lane & 15;
  v8f c[MT] = {};
  wmma_mt(a, D, Wo, D, col0, c);
  const float bias = bo[col0 + lm];
#pragma unroll
  for (int m = 0; m < MT; ++m) {
    const size_t base = (size_t)(row0 + m * 16 + (half << 3)) * D + col0 + lm;
    for (int r = 0; r < 8; ++r)
      h1[base + (size_t)r * D] = c[m][r] + bias + x[base + (size_t)r * D];
  }
}

// ---------------------------------------------------------------------------
// BatchNorm: per-channel sum/sumsq, finalize, fused scale/shift apply.
// ---------------------------------------------------------------------------
__global__ void k_bnstats(const float* __restrict__ h, float* __restrict__ stats,
                          int n) {
  const int c = threadIdx.x;                    // blockDim.x == D
  float s = 0.f, s2 = 0.f;
  for (int r = blockIdx.x; r < n; r += gridDim.x) {
    const float v = h[(size_t)r * D + c];
    s += v;
    s2 += v * v;
  }
  atomicAdd(stats + c, s);
  atomicAdd(stats + D + c, s2);
}

__global__ void k_bnfinal(const float* __restrict__ stats, const float* __restrict__ g,
                          const float* __restrict__ be, float* __restrict__ ss,
                          int n) {
  const int c = threadIdx.x;
  const float inv_n = 1.f / (float)n;
  const float mu = stats[c] * inv_n;
  const float var = stats[D + c] * inv_n - mu * mu;
  const float sc = g[c] * rsqrtf(var + EPSV);
  ss[c] = sc;
  ss[D + c] = be[c] - mu * sc;
}

__global__ void k_bnapply(const float* __restrict__ h, const float* __restrict__ ss,
                          float* __restrict__ out, int nd4) {
  const int i = blockIdx.x * blockDim.x + threadIdx.x;
  if (i >= nd4) return;
  float4 v = ((const float4*)h)[i];
  const int c = (i & (D / 4 - 1)) << 2;
  v.x = v.x * ss[c + 0] + ss[D + c + 0];
  v.y = v.y * ss[c + 1] + ss[D + c + 1];
  v.z = v.z * ss[c + 2] + ss[D + c + 2];
  v.w = v.w * ss[c + 3] + ss[D + c + 3];
  ((float4*)out)[i] = v;
}

// ---------------------------------------------------------------------------
// Stage 5: y = relu(hbn1 @ W1 + b1)   (K=128, 256 output cols: 2 tiles/wave)
// ---------------------------------------------------------------------------
__global__ void k_ffn1(const float* __restrict__ hbn, const float* __restrict__ W1,
                       const float* __restrict__ b1, float* __restrict__ y) {
  __shared__ float a[MT * 16 * LDA];
  const int row0 = blockIdx.x * (MT * 16);
  load_atile(a, hbn + (size_t)row0 * D, D);

  const int wave = threadIdx.x >> 5;
  const int lane = threadIdx.x & 31, half = lane >> 4, lm = lane & 15;
  for (int t = wave; t < 16; t += 8) {
    const int col0 = t * 16;
    v8f c[MT] = {};
    wmma_mt(a, D, W1, 2 * D, col0, c);
    const float bias = b1[col0 + lm];
#pragma unroll
    for (int m = 0; m < MT; ++m) {
      float* o = y + (size_t)(row0 + m * 16 + (half << 3)) * (2 * D) + col0 + lm;
      for (int r = 0; r < 8; ++r) {
        const float v = c[m][r] + bias;
        o[(size_t)r * (2 * D)] = v > 0.f ? v : 0.f;
      }
    }
  }
}

// ---------------------------------------------------------------------------
// Stage 6: h2 = hbn1 + y @ W2 + b2   (K=256, processed as two K=128 chunks so
// the padded LDS A-tile stays at 80x132 floats)
// ---------------------------------------------------------------------------
__global__ void k_ffn2(const float* __restrict__ y, const float* __restrict__ W2,
                       const float* __restrict__ b2, const float* __restrict__ hbn,
                       float* __restrict__ h2) {
  __shared__ float a[MT * 16 * LDA];
  const int row0 = blockIdx.x * (MT * 16);
  const int wave = threadIdx.x >> 5;
  const int col0 = wave * 16;
  const int lane = threadIdx.x & 31, half = lane >> 4, lm = lane & 15;

  v8f c[MT] = {};
  for (int kc = 0; kc < 2 * D; kc += D) {
    __syncthreads();   // protect LDS from previous chunk's readers
    for (int i = threadIdx.x; i < MT * 16 * (D / 4); i += blockDim.x) {
      const int r  = i >> 5;
      const int c4 = (i & 31) << 2;
      copy16(a + r * LDA + c4, y + (size_t)(row0 + r) * (2 * D) + kc + c4);
    }
    copy_fence();
    wmma_mt(a, D, W2 + (size_t)kc * D, D, col0, c);
  }

  const float bias = b2[col0 + lm];
#pragma unroll
  for (int m = 0; m < MT; ++m) {
    const size_t base = (size_t)(row0 + m * 16 + (half << 3)) * D + col0 + lm;
    for (int r = 0; r < 8; ++r)
      h2[base + (size_t)r * D] = c[m][r] + bias + hbn[base + (size_t)r * D];
  }
}

__global__ void k_fill0(float* __restrict__ p, size_t n) {
  const size_t i = (size_t)blockIdx.x * blockDim.x + threadIdx.x;
  if (i < n) p[i] = 0.f;
}

// ---------------------------------------------------------------------------
extern "C" void kernel_launch(void* const* d_in, const int* in_sizes, int n_in,
                              void* d_out, int out_size, void* d_ws, size_t ws_size,
                              hipStream_t stream) {
  const float* x  = (const float*)d_in[0];
  const long long* ei = (const long long*)d_in[1];
  const float* Wq = (const float*)d_in[2];
  const float* Wk = (const float*)d_in[3];
  const float* Wv = (const float*)d_in[4];
  const float* Wo = (const float*)d_in[5];
  const float* bo = (const float*)d_in[6];
  const float* W1 = (const float*)d_in[7];
  const float* b1 = (const float*)d_in[8];
  const float* W2 = (const float*)d_in[9];
  const float* b2 = (const float*)d_in[10];
  const float* g1 = (const float*)d_in[11];
  const float* be1 = (const float*)d_in[12];
  const float* g2 = (const float*)d_in[13];
  const float* be2 = (const float*)d_in[14];

  const int N = in_sizes[0] / D;       // 50000 (== 625 * 80)
  const int E = in_sizes[1] / 2;       // 800000
  const long long* src = ei;           // edge_index[0]
  const long long* dst = ei + E;       // edge_index[1]

  const size_t ND = (size_t)N * D;
  float* ws = (float*)d_ws;
  float* Q      = ws; ws += ND;
  float* Km     = ws; ws += ND;
  float* Vm     = ws; ws += ND;
  float* ex     = ws; ws += (size_t)E * HH;
  float* denom  = ws; ws += (size_t)N * HH;
  float* att    = ws; ws += ND;
  float* y      = ws; ws += 2 * ND;
  float* stats1 = ws; ws += 2 * D;
  float* ss1    = ws; ws += 2 * D;
  float* stats2 = ws; ws += 2 * D;
  float* ss2    = ws; ws += 2 * D;
  // reuse (each source buffer is dead before its reuse begins):
  float* h1   = Q;    // Q last read in k_scores
  float* hbn1 = Km;   // K last read in k_scores
  float* h2   = att;  // att last read in k_oproj

  const int rowBlocks = N / (MT * 16);   // 625

  {
    const size_t n = (size_t)N * HH;
    k_fill0<<<(unsigned)((n + 255) / 256), 256, 0, stream>>>(denom, n);
  }
  k_fill0<<<(unsigned)((ND + 255) / 256), 256, 0, stream>>>(att, ND);
  k_fill0<<<1, 256, 0, stream>>>(stats1, (size_t)(2 * D));
  k_fill0<<<1, 256, 0, stream>>>(stats2, (size_t)(2 * D));

  k_qkv<<<rowBlocks, 256, 0, stream>>>(x, Wq, Wk, Wv, Q, Km, Vm);

  {
    const long long n = (long long)E * HH;
    k_scores<<<(unsigned)((n + 255) / 256), 256, 0, stream>>>(Q, Km, src, dst, ex,
                                                              denom, E);
  }
  {
    const long long n = (long long)E * D;
    k_aggregate<<<(unsigned)((n + 255) / 256), 256, 0, stream>>>(Vm, ex, denom, src,
                                                                 dst, att, E);
  }

  k_oproj<<<rowBlocks, 256, 0, stream>>>(att, Wo, bo, x, h1);

  k_bnstats<<<512, D, 0, stream>>>(h1, stats1, N);
  k_bnfinal<<<1, D, 0, stream>>>(stats1, g1, be1, ss1, N);
  {
    const int n4 = (int)(ND / 4);
    k_bnapply<<<(n4 + 255) / 256, 256, 0, stream>>>(h1, ss1, hbn1, n4);
  }

  k_ffn1<<<rowBlocks, 256, 0, stream>>>(hbn1, W1, b1, y);
  k_ffn2<<<rowBlocks, 256, 0, stream>>>(y, W2, b2, hbn1, h2);

  k_bnstats<<<512, D, 0, stream>>>(h2, stats2, N);
  k_bnfinal<<<1, D, 0, stream>>>(stats2, g2, be2, ss2, N);
  {
    const int n4 = (int)(ND / 4);
    k_bnapply<<<(n4 + 255) / 256, 256, 0, stream>>>(h2, ss2, (float*)d_out, n4);
  }
}